// FourierController_78486232367440
// MI455X (gfx1250) — compile-verified
//
#include <hip/hip_runtime.h>
#include <math.h>

typedef __attribute__((ext_vector_type(2))) float v2f;
typedef __attribute__((ext_vector_type(8))) float v8f;

#define Bb 4
#define Tt 1024
#define Dd 256
#define Mm 32
#define NROWS (Bb * Tt)

__device__ __forceinline__ float gelu_exact(float x) {
    // jax.nn.gelu(approximate=False) = 0.5*x*(1+erf(x/sqrt(2)))
    return 0.5f * x * (1.0f + erff(x * 0.70710678118654752440f));
}

// ---------------------------------------------------------------------------
// LayerNorm over D=256: one row per 256-thread block (8 wave32s on CDNA5).
// ---------------------------------------------------------------------------
__global__ void ln_kernel(const float* __restrict__ in,
                          const float* __restrict__ gamma,
                          const float* __restrict__ beta,
                          float* __restrict__ out) {
    __shared__ float red[256];
    const int row = blockIdx.x;
    const int tid = threadIdx.x;
    float v = in[(size_t)row * Dd + tid];
    red[tid] = v;
    __syncthreads();
    for (int s = 128; s > 0; s >>= 1) {
        if (tid < s) red[tid] += red[tid + s];
        __syncthreads();
    }
    float mu = red[0] * (1.0f / (float)Dd);
    __syncthreads();
    float dv = v - mu;
    red[tid] = dv * dv;
    __syncthreads();
    for (int s = 128; s > 0; s >>= 1) {
        if (tid < s) red[tid] += red[tid + s];
        __syncthreads();
    }
    float var = red[0] * (1.0f / (float)Dd);
    out[(size_t)row * Dd + tid] = dv * rsqrtf(var + 1e-5f) * gamma[tid] + beta[tid];
}

// ---------------------------------------------------------------------------
// Fold spectral weights with the IDFT vector:
//   c[m,d] = sum_n (w_re + i*w_im)[m,n,d] * idft[n]
//   idft[n] = (cos(2*pi*n/T) - i*sin(2*pi*n/T)) * (n==0 ? 1 : 2)/T
// ---------------------------------------------------------------------------
__global__ void cfold_kernel(const float* __restrict__ w_re,
                             const float* __restrict__ w_im,
                             float* __restrict__ c_re,
                             float* __restrict__ c_im) {
    const int idx = blockIdx.x * blockDim.x + threadIdx.x;  // m*D + d
    if (idx >= Mm * Dd) return;
    const int m = idx / Dd;
    const int d = idx - m * Dd;
    const float w = 6.28318530717958647692f / (float)Tt;
    float cre = 0.f, cim = 0.f;
#pragma unroll 4
    for (int n = 0; n < Mm; ++n) {
        float sc = (n == 0 ? 1.0f : 2.0f) / (float)Tt;
        float sn, cs;
        __sincosf(w * (float)n, &sn, &cs);
        float ire = cs * sc;
        float iim = -sn * sc;
        float wr = w_re[(size_t)(m * Mm + n) * Dd + d];
        float wi = w_im[(size_t)(m * Mm + n) * Dd + d];
        cre += wr * ire - wi * iim;
        cim += wr * iim + wi * ire;
    }
    c_re[idx] = cre;
    c_im[idx] = cim;
}

// ---------------------------------------------------------------------------
// Causal sliding-DFT scan. One wave32 per (b,d) pair; lane == mode m.
//   S_m(t) = S_m(t-1) + h[b,t,d] * exp(-i*2*pi*m*t/T)      (exact LDS table)
//   y[b,t,d] = Re( sum_m exp(+i*2*pi*m*(t+1)/T) * S_m(t) * c[m,d] )
// Then epilogue: resid = gelu(y) + x.
// ---------------------------------------------------------------------------
__global__ void specscan_kernel(const float* __restrict__ h,
                                const float* __restrict__ c_re,
                                const float* __restrict__ c_im,
                                const float* __restrict__ x,
                                float* __restrict__ resid_out) {
    __shared__ float tab_c[Tt];
    __shared__ float tab_s[Tt];
    for (int i = threadIdx.x; i < Tt; i += blockDim.x) {
        float sn, cs;
        __sincosf(6.28318530717958647692f * (float)i / (float)Tt, &sn, &cs);
        tab_c[i] = cs;
        tab_s[i] = sn;
    }
    __syncthreads();

    const int wave = threadIdx.x >> 5;
    const int lane = threadIdx.x & 31;
    const int pair = blockIdx.x * (blockDim.x >> 5) + wave;  // b*D + d
    const int b = pair / Dd;
    const int d = pair - b * Dd;
    const int m = lane;

    const float cr = c_re[m * Dd + d];
    const float ci = c_im[m * Dd + d];
    float Sre = 0.f, Sim = 0.f;

    const float* hrow = h + (size_t)b * Tt * Dd + d;
    const float* xrow = x + (size_t)b * Tt * Dd + d;
    float* orow = resid_out + (size_t)b * Tt * Dd + d;

    for (int t = 0; t < Tt; ++t) {
        float hv = hrow[(size_t)t * Dd];
        int k1 = (m * t) & (Tt - 1);          // exp(-i th): (cos, -sin)
        Sre = fmaf(hv, tab_c[k1], Sre);
        Sim = fmaf(-hv, tab_s[k1], Sim);
        float Pre = Sre * cr - Sim * ci;
        float Pim = Sre * ci + Sim * cr;
        int k2 = (m * (t + 1)) & (Tt - 1);    // exp(+i th): (cos, +sin)
        float contrib = tab_c[k2] * Pre - tab_s[k2] * Pim;
        // sum over the 32 modes (wave32 butterfly)
        contrib += __shfl_xor(contrib, 16, 32);
        contrib += __shfl_xor(contrib, 8, 32);
        contrib += __shfl_xor(contrib, 4, 32);
        contrib += __shfl_xor(contrib, 2, 32);
        contrib += __shfl_xor(contrib, 1, 32);
        if (lane == 0) {
            orow[(size_t)t * Dd] = gelu_exact(contrib) + xrow[(size_t)t * Dd];
        }
    }
}

// ---------------------------------------------------------------------------
// f32 GEMM on the matrix pipe: out[row,j] = sum_k A[row,k]*W[j,k] + bias[j]
// via V_WMMA_F32_16X16X4_F32.
// Each wave owns a 16x64 output strip (four 16x16 N-tiles sharing one A
// fragment -> 1.25 fragment loads per wmma) and is software-pipelined one
// K-step ahead so global loads overlap the previous 4 wmmas. Branch-free
// inner loop. All 32 lanes active (WMMA requires EXEC all-1s).
// A/B fragment layout (16x4 f32): lanes 0-15 hold K=0,1; lanes 16-31 K=2,3.
// mode 0: epilogue = exact GELU.  mode 1: epilogue = + resid.
// ---------------------------------------------------------------------------
__global__ void gemm_wmma_kernel(const float* __restrict__ A,
                                 const float* __restrict__ W,
                                 const float* __restrict__ bias,
                                 const float* __restrict__ resid,
                                 float* __restrict__ out,
                                 int mode) {
    const int N_GROUPS = Dd / 64;  // 4 column groups of 64
    const int wave = threadIdx.x >> 5;
    const int lane = threadIdx.x & 31;
    const int wtile = blockIdx.x * (blockDim.x >> 5) + wave;
    const int tile_m = wtile / N_GROUPS;
    const int ng = wtile - tile_m * N_GROUPS;
    const int half = lane >> 4;  // 0 or 1 -> K offset 0 or 2
    const int lid = lane & 15;

    const float* arow = A + (size_t)(tile_m * 16 + lid) * Dd + (half << 1);
    const float* brow = W + (size_t)(ng * 64 + lid) * Dd + (half << 1);
    const size_t bstep = (size_t)16 * Dd;  // next 16x16 N-tile of W

    // warm caches for the strip (branch-free, outside the loop)
    __builtin_prefetch(arow, 0, 1);
    __builtin_prefetch(brow, 0, 1);
    __builtin_prefetch(brow + 2 * bstep, 0, 1);

    v8f acc0 = {}, acc1 = {}, acc2 = {}, acc3 = {};

    // software pipeline: fragments for K-step k are loaded at step k-4
    v2f a_c  = *(const v2f*)(arow);
    v2f b0_c = *(const v2f*)(brow);
    v2f b1_c = *(const v2f*)(brow + bstep);
    v2f b2_c = *(const v2f*)(brow + 2 * bstep);
    v2f b3_c = *(const v2f*)(brow + 3 * bstep);

#pragma unroll 4
    for (int k = 0; k < Dd - 4; k += 4) {
        v2f a_n  = *(const v2f*)(arow + k + 4);
        v2f b0_n = *(const v2f*)(brow + k + 4);
        v2f b1_n = *(const v2f*)(brow + bstep + k + 4);
        v2f b2_n = *(const v2f*)(brow + 2 * bstep + k + 4);
        v2f b3_n = *(const v2f*)(brow + 3 * bstep + k + 4);
        acc0 = __builtin_amdgcn_wmma_f32_16x16x4_f32(false, a_c, false, b0_c,
                                                     (short)0, acc0, false, false);
        acc1 = __builtin_amdgcn_wmma_f32_16x16x4_f32(false, a_c, false, b1_c,
                                                     (short)0, acc1, false, false);
        acc2 = __builtin_amdgcn_wmma_f32_16x16x4_f32(false, a_c, false, b2_c,
                                                     (short)0, acc2, false, false);
        acc3 = __builtin_amdgcn_wmma_f32_16x16x4_f32(false, a_c, false, b3_c,
                                                     (short)0, acc3, false, false);
        a_c = a_n; b0_c = b0_n; b1_c = b1_n; b2_c = b2_n; b3_c = b3_n;
    }
    // peeled last K-step
    acc0 = __builtin_amdgcn_wmma_f32_16x16x4_f32(false, a_c, false, b0_c,
                                                 (short)0, acc0, false, false);
    acc1 = __builtin_amdgcn_wmma_f32_16x16x4_f32(false, a_c, false, b1_c,
                                                 (short)0, acc1, false, false);
    acc2 = __builtin_amdgcn_wmma_f32_16x16x4_f32(false, a_c, false, b2_c,
                                                 (short)0, acc2, false, false);
    acc3 = __builtin_amdgcn_wmma_f32_16x16x4_f32(false, a_c, false, b3_c,
                                                 (short)0, acc3, false, false);

    // epilogue: bias + (GELU | resid), 16x64 strip
    v8f accs[4] = {acc0, acc1, acc2, acc3};
#pragma unroll
    for (int j = 0; j < 4; ++j) {
        const int col = ng * 64 + j * 16 + lid;
        const float bv = bias[col];
#pragma unroll
        for (int r = 0; r < 8; ++r) {
            // C/D layout: VGPR r -> M = r (lanes 0-15) or r+8 (lanes 16-31)
            int row = tile_m * 16 + r + (half << 3);
            float v = accs[j][r] + bv;
            if (mode == 0) {
                v = gelu_exact(v);
            } else {
                v += resid[(size_t)row * Dd + col];
            }
            out[(size_t)row * Dd + col] = v;
        }
    }
}

// ---------------------------------------------------------------------------
extern "C" void kernel_launch(void* const* d_in, const int* in_sizes, int n_in,
                              void* d_out, int out_size, void* d_ws, size_t ws_size,
                              hipStream_t stream) {
    const float* x     = (const float*)d_in[0];
    const float* w_re  = (const float*)d_in[1];
    const float* w_im  = (const float*)d_in[2];
    const float* ln1_g = (const float*)d_in[3];
    const float* ln1_b = (const float*)d_in[4];
    const float* ln2_g = (const float*)d_in[5];
    const float* ln2_b = (const float*)d_in[6];
    const float* fc1_w = (const float*)d_in[7];
    const float* fc1_b = (const float*)d_in[8];
    const float* fc2_w = (const float*)d_in[9];
    const float* fc2_b = (const float*)d_in[10];

    float* ws    = (float*)d_ws;
    float* h     = ws;                      // B*T*D = 1,048,576 floats
    float* resid = ws + 1048576;            // B*T*D
    float* g     = ws + 2097152;            // B*T*D
    float* a1    = ws + 3145728;            // B*T*D
    float* c_re  = ws + 4194304;            // M*D = 8192
    float* c_im  = ws + 4194304 + 8192;     // M*D
    float* outp  = (float*)d_out;

    // 1) h = layernorm1(x)
    ln_kernel<<<NROWS, 256, 0, stream>>>(x, ln1_g, ln1_b, h);
    // 2) fold spectral weights with idft -> c[m,d]
    cfold_kernel<<<(Mm * Dd + 255) / 256, 256, 0, stream>>>(w_re, w_im, c_re, c_im);
    // 3) causal sliding-DFT scan + gelu + residual -> resid
    specscan_kernel<<<(Bb * Dd) / 8, 256, 0, stream>>>(h, c_re, c_im, x, resid);
    // 4) g = layernorm2(resid)
    ln_kernel<<<NROWS, 256, 0, stream>>>(resid, ln2_g, ln2_b, g);
    // 5) a1 = gelu(g @ fc1_w^T + fc1_b)          (WMMA f32, 16x64 strips)
    gemm_wmma_kernel<<<(NROWS / 16) * (Dd / 64) / 8, 256, 0, stream>>>(
        g, fc1_w, fc1_b, nullptr, a1, 0);
    // 6) out = a1 @ fc2_w^T + fc2_b + resid      (WMMA f32, 16x64 strips)
    gemm_wmma_kernel<<<(NROWS / 16) * (Dd / 64) / 8, 256, 0, stream>>>(
        a1, fc2_w, fc2_b, resid, outp, 1);
}